// ABDMBR_74655121539772
// MI455X (gfx1250) — compile-verified
//
#include <hip/hip_runtime.h>
#include <math.h>

#define N_USERS 60000
#define N_ITEMS 40000
#define DIMS    64
#define NB      3
#define NE      600000
#define NBATCH  2048
#define NNODES  (N_USERS + 1 + N_ITEMS + 1)   // 100002
#define UOFF    (N_USERS + 1)                  // 60001
#define NND     ((long)NNODES * DIMS)

typedef __attribute__((ext_vector_type(2))) float v2f;
typedef __attribute__((ext_vector_type(8))) float v8f;

// ---------------- utility ----------------
__device__ inline float wred(float v) {
#pragma unroll
  for (int m = 16; m >= 1; m >>= 1) v += __shfl_xor(v, m, 32);
  return v;
}

__global__ void zero_kernel(float* __restrict__ p, long n) {
  long i = (long)blockIdx.x * blockDim.x + threadIdx.x;
  if (i < n) p[i] = 0.0f;
}

// ---------------- degree counts (exact integral fp32 sums) ----------------
__global__ void deg_kernel(const int* __restrict__ eu, const int* __restrict__ ei,
                           float* __restrict__ deg, int ne) {
  int e = blockIdx.x * blockDim.x + threadIdx.x;
  if (e >= ne) return;
  int u = eu[e];
  int v = ei[e] + UOFF;
  atomicAdd(&deg[u], 1.0f);
  atomicAdd(&deg[v], 1.0f);
}

// ---------------- one propagation layer: nxt += norm * cur[src] scattered to dst
// cur==nullptr => read from concat(user_emb, item_emb)
__global__ void scatter_kernel(const float* __restrict__ cur,
                               const float* __restrict__ ue, const float* __restrict__ ie,
                               float* __restrict__ nxt,
                               const int* __restrict__ eu, const int* __restrict__ ei,
                               const float* __restrict__ deg, int ne) {
  long t = (long)blockIdx.x * blockDim.x + threadIdx.x;
  long total = (long)ne * 16;
  if (t >= total) return;
  int e = (int)(t >> 4);
  int q = (int)(t & 15) * 4;
  int u = eu[e];
  int v = ei[e] + UOFF;
  float du = deg[u], dv = deg[v];
  float nrm = (1.0f / sqrtf(fmaxf(du, 1.0f))) * (1.0f / sqrtf(fmaxf(dv, 1.0f)));
  const float* rowu;
  const float* rowv;
  if (cur) {
    rowu = cur + (long)u * DIMS;
    rowv = cur + (long)v * DIMS;
  } else {
    rowu = (u < UOFF) ? (ue + (long)u * DIMS) : (ie + (long)(u - UOFF) * DIMS);
    rowv = ie + (long)(v - UOFF) * DIMS;
  }
  float4 cu = *(const float4*)(rowu + q);
  float4 cv = *(const float4*)(rowv + q);
  float* nv = nxt + (long)v * DIMS + q;
  float* nu = nxt + (long)u * DIMS + q;
  atomicAdd(&nv[0], cu.x * nrm); atomicAdd(&nv[1], cu.y * nrm);
  atomicAdd(&nv[2], cu.z * nrm); atomicAdd(&nv[3], cu.w * nrm);
  atomicAdd(&nu[0], cv.x * nrm); atomicAdd(&nu[1], cv.y * nrm);
  atomicAdd(&nu[2], cv.z * nrm); atomicAdd(&nu[3], cv.w * nrm);
}

// ---------------- all_emb = (emb0 + n1 + n2)/3 (global run) ----------------
__global__ void combine_kernel(const float* __restrict__ ue, const float* __restrict__ ie,
                               const float* __restrict__ n1, const float* __restrict__ n2,
                               float* __restrict__ outp) {
  long i = (long)blockIdx.x * blockDim.x + threadIdx.x;
  if (i >= NND) return;
  long node = i >> 6;
  float e0 = (node < UOFF) ? ue[i] : ie[i - (long)UOFF * DIMS];
  outp[i] = (e0 + n1[i] + n2[i]) * (1.0f / 3.0f);
}

// ---------------- gather branch-j output rows at batch users/items -------------
// Ubuf[j][b][s][d], Ibuf[j][b][s][k][d]; branch output row = (all_emb+n1+n2)/3
__global__ void extract_kernel(int j, const float* __restrict__ ae,
                               const float* __restrict__ n1, const float* __restrict__ n2,
                               const int* __restrict__ bd,
                               float* __restrict__ Ub, float* __restrict__ Ib) {
  long idx = (long)blockIdx.x * blockDim.x + threadIdx.x;
  const long TOT = (long)NB * NBATCH * 3 * DIMS;
  if (idx >= TOT) return;
  int d = (int)(idx & 63);
  long rest = idx >> 6;               // ((b*NBATCH)+s)*3 + c
  int c = (int)(rest % 3);
  long bs = rest / 3;
  int s = (int)(bs % NBATCH);
  int b = (int)(bs / NBATCH);
  int raw = bd[((long)s * NB + b) * 3 + c];
  long node = (c == 0) ? (long)raw : (long)(UOFF + raw);
  long o = node * DIMS + d;
  float val = (ae[o] + n1[o] + n2[o]) * (1.0f / 3.0f);
  if (c == 0)
    Ub[(((long)j * NB + b) * NBATCH + s) * DIMS + d] = val;
  else
    Ib[((((long)j * NB + b) * NBATCH + s) * 2 + (c - 1)) * DIMS + d] = val;
}

// ---------------- item_final at needed (b,s,k) occurrences ----------------
__global__ void itemfinal_kernel(const int* __restrict__ bd,
                                 const float* __restrict__ dg0, const float* __restrict__ dg1,
                                 const float* __restrict__ dg2, const float* __restrict__ W,
                                 const float* __restrict__ Ib, float* __restrict__ IPN) {
  long idx = (long)blockIdx.x * blockDim.x + threadIdx.x;
  const long TOT = (long)NB * NBATCH * 2 * DIMS;
  if (idx >= TOT) return;
  int d = (int)(idx & 63);
  long rest = idx >> 6;               // ((b*NBATCH)+s)*2 + k
  int k = (int)(rest & 1);
  long bs = rest >> 1;
  int s = (int)(bs % NBATCH);
  int b = (int)(bs / NBATCH);
  int item = bd[((long)s * NB + b) * 3 + 1 + k];
  float w0 = dg0[UOFF + item] * W[0];
  float w1 = dg1[UOFF + item] * W[1];
  float w2 = dg2[UOFF + item] * W[2];
  float ws = w0 + w1 + w2 + 1e-8f;
  long base = (((long)b * NBATCH + s) * 2 + k) * DIMS + d;
  long stride = (long)NB * NBATCH * 2 * DIMS;   // per-j stride in Ibuf
  long ib = ((((long)0 * NB + b) * NBATCH + s) * 2 + k) * DIMS + d;
  float v = Ib[ib] * w0 + Ib[ib + stride] * w1 + Ib[ib + 2 * stride] * w2;
  IPN[base] = v / ws;
}

// ---------------- mutual attention: one wave per (b,s) pair ----------------
__global__ __launch_bounds__(256) void attention_kernel(const float* __restrict__ Ub,
                                                        float* __restrict__ UF) {
  int pair = blockIdx.x * 8 + (threadIdx.x >> 5);   // 6144 pairs
  int l = threadIdx.x & 31;
  int b = pair / NBATCH;
  int s = pair % NBATCH;
  float fa[3], fb[3];
#pragma unroll
  for (int j = 0; j < 3; j++) {
    const float* row = Ub + (((long)j * NB + b) * NBATCH + s) * DIMS;
    fa[j] = row[l];
    fb[j] = row[l + 32];
  }
  float s00 = wred(fa[0] * fa[0] + fb[0] * fb[0]);
  float s01 = wred(fa[0] * fa[1] + fb[0] * fb[1]);
  float s02 = wred(fa[0] * fa[2] + fb[0] * fb[2]);
  float s11 = wred(fa[1] * fa[1] + fb[1] * fb[1]);
  float s12 = wred(fa[1] * fa[2] + fb[1] * fb[2]);
  float s22 = wred(fa[2] * fa[2] + fb[2] * fb[2]);
  float last[3] = {s02, s12, s22};
  float p0[3] = {s00, s01, s02};
  float p1[3] = {s01, s11, s12};
  float c0[3], c1[3], sc[3];
#pragma unroll
  for (int j = 0; j < 3; j++) {
    float l2 = last[j] * last[j];
    float inv = 1.0f / (l2 + 1e-12f);
    c0[j] = last[j] * p0[j] * last[j] * inv;
    c1[j] = last[j] * p1[j] * last[j] * inv;
    sc[j] = c0[j] + c1[j] + last[j];
  }
  float row[3];
#pragma unroll
  for (int j = 0; j < 3; j++)
    row[j] = (b == 0) ? c0[j] : ((b == 1) ? c1[j] : sc[j]);
  // softmax(row / sqrt(64))
  float x0 = row[0] * 0.125f, x1 = row[1] * 0.125f, x2 = row[2] * 0.125f;
  float m = fmaxf(x0, fmaxf(x1, x2));
  float e0 = expf(x0 - m), e1 = expf(x1 - m), e2 = expf(x2 - m);
  float inv = 1.0f / (e0 + e1 + e2);
  float a0 = e0 * inv, a1 = e1 * inv, a2 = e2 * inv;
  float oa = a0 * fa[0] + a1 * fa[1] + a2 * fa[2];
  float ob = a0 * fb[0] + a1 * fb[1] + a2 * fb[2];
  UF[(long)pair * DIMS + l] = oa;
  UF[(long)pair * DIMS + l + 32] = ob;
}

// ---------------- BPR loss: batched dots as diag of WMMA fp32 GEMM ----------
__device__ inline float sel8(v8f c, int i) {
  float x = c[0];
  if (i == 1) x = c[1];
  if (i == 2) x = c[2];
  if (i == 3) x = c[3];
  if (i == 4) x = c[4];
  if (i == 5) x = c[5];
  if (i == 6) x = c[6];
  if (i == 7) x = c[7];
  return x;
}

__global__ __launch_bounds__(32) void bpr_wmma_kernel(const float* __restrict__ UF,
                                                      const float* __restrict__ IPN,
                                                      float* __restrict__ acc) {
  int L = threadIdx.x;       // 0..31, full wave, EXEC all ones for WMMA
  int r = L & 15;
  int h = L >> 4;
  long p = (long)blockIdx.x * 16 + r;      // pair id
  const float* arow = UF + p * DIMS;
  const float* prow = IPN + (p * 2 + 0) * DIMS;
  const float* nrow = IPN + (p * 2 + 1) * DIMS;
  v8f dpos = {};
  v8f dneg = {};
#pragma unroll
  for (int c = 0; c < 16; c++) {
    int o = 4 * c + 2 * h;
    v2f a, bp, bn;
    a[0] = arow[o];  a[1] = arow[o + 1];
    bp[0] = prow[o]; bp[1] = prow[o + 1];
    bn[0] = nrow[o]; bn[1] = nrow[o + 1];
    dpos = __builtin_amdgcn_wmma_f32_16x16x4_f32(false, a, false, bp, (short)0, dpos, false, false);
    dneg = __builtin_amdgcn_wmma_f32_16x16x4_f32(false, a, false, bn, (short)0, dneg, false, false);
  }
  // diagonal: pair n<8 lives in lane n, VGPR n; pair n>=8 in lane n+16, VGPR n-8
  bool owner = (L < 8) || (L >= 24);
  int idx = (L < 8) ? L : (L - 24);
  float sp = sel8(dpos, idx);
  float sn = sel8(dneg, idx);
  if (owner) {
    float diff = sp - sn;
    float sig = 1.0f / (1.0f + expf(-diff));
    float lv = logf(1e-10f + sig);
    atomicAdd(acc, -lv * (1.0f / (float)NBATCH));
  }
}

// ---------------- sum of squares reduction ----------------
__global__ void sqsum_kernel(const float* __restrict__ x, long n, float* __restrict__ out) {
  long i0 = (long)blockIdx.x * blockDim.x + threadIdx.x;
  long stride = (long)gridDim.x * blockDim.x;
  float a = 0.0f;
  for (long i = i0; i < n; i += stride) {
    float v = x[i];
    a += v * v;
  }
  a = wred(a);
  if ((threadIdx.x & 31) == 0) atomicAdd(out, a);
}

__global__ void finalize_kernel(const float* __restrict__ acc, float* __restrict__ out) {
  if (blockIdx.x == 0 && threadIdx.x == 0) {
    float emb_loss = (sqrtf(acc[1]) + sqrtf(acc[2])) / (float)(N_ITEMS + 1);
    out[0] = acc[0] + 0.001f * emb_loss;
  }
}

// ---------------- host orchestration ----------------
static inline unsigned nblk(long n) { return (unsigned)((n + 255) / 256); }

extern "C" void kernel_launch(void* const* d_in, const int* in_sizes, int n_in,
                              void* d_out, int out_size, void* d_ws, size_t ws_size,
                              hipStream_t stream) {
  const float* ue = (const float*)d_in[0];
  const float* ie = (const float*)d_in[1];
  const float* W  = (const float*)d_in[2];
  const int*   eu = (const int*)d_in[3];
  const int*   ei = (const int*)d_in[4];
  const int*   bd = (const int*)d_in[5];
  float* out = (float*)d_out;

  float* ws = (float*)d_ws;
  float* all_emb = ws;
  float* n1 = all_emb + NND;
  float* n2 = n1 + NND;
  float* deg_g = n2 + NND;
  float* degb0 = deg_g + NNODES;
  float* degb1 = degb0 + NNODES;
  float* degb2 = degb1 + NNODES;
  float* Ubuf = degb2 + NNODES;                              // [3][3][2048][64]
  float* Ibuf = Ubuf + (long)NB * NB * NBATCH * DIMS;        // [3][3][2048][2][64]
  float* UF   = Ibuf + (long)NB * NB * NBATCH * 2 * DIMS;    // [3][2048][64]
  float* IPN  = UF + (long)NB * NBATCH * DIMS;               // [3][2048][2][64]
  float* acc  = IPN + (long)NB * NBATCH * 2 * DIMS;          // [loss, ssu, ssi]
  float* degb[3] = {degb0, degb1, degb2};

  // init
  zero_kernel<<<nblk(NNODES), 256, 0, stream>>>(deg_g, NNODES);
  for (int b = 0; b < 3; b++)
    zero_kernel<<<nblk(NNODES), 256, 0, stream>>>(degb[b], NNODES);
  zero_kernel<<<1, 32, 0, stream>>>(acc, 8);

  // degrees
  deg_kernel<<<nblk((long)3 * NE), 256, 0, stream>>>(eu, ei, deg_g, 3 * NE);
  for (int b = 0; b < 3; b++)
    deg_kernel<<<nblk(NE), 256, 0, stream>>>(eu + (long)b * NE, ei + (long)b * NE, degb[b], NE);

  // global LightGCN (2 layers) from concat(user_emb,item_emb)
  zero_kernel<<<nblk(NND), 256, 0, stream>>>(n1, NND);
  scatter_kernel<<<nblk((long)3 * NE * 16), 256, 0, stream>>>(nullptr, ue, ie, n1, eu, ei, deg_g, 3 * NE);
  zero_kernel<<<nblk(NND), 256, 0, stream>>>(n2, NND);
  scatter_kernel<<<nblk((long)3 * NE * 16), 256, 0, stream>>>(n1, nullptr, nullptr, n2, eu, ei, deg_g, 3 * NE);
  combine_kernel<<<nblk(NND), 256, 0, stream>>>(ue, ie, n1, n2, all_emb);

  // branch LightGCNs; extract only the batch-needed rows of each branch output
  for (int j = 0; j < 3; j++) {
    const int* beu = eu + (long)j * NE;
    const int* bei = ei + (long)j * NE;
    zero_kernel<<<nblk(NND), 256, 0, stream>>>(n1, NND);
    scatter_kernel<<<nblk((long)NE * 16), 256, 0, stream>>>(all_emb, nullptr, nullptr, n1, beu, bei, degb[j], NE);
    zero_kernel<<<nblk(NND), 256, 0, stream>>>(n2, NND);
    scatter_kernel<<<nblk((long)NE * 16), 256, 0, stream>>>(n1, nullptr, nullptr, n2, beu, bei, degb[j], NE);
    extract_kernel<<<nblk((long)NB * NBATCH * 3 * DIMS), 256, 0, stream>>>(j, all_emb, n1, n2, bd, Ubuf, Ibuf);
  }

  // item_final at batch items, attention at batch users, BPR loss via WMMA
  itemfinal_kernel<<<nblk((long)NB * NBATCH * 2 * DIMS), 256, 0, stream>>>(bd, degb0, degb1, degb2, W, Ibuf, IPN);
  attention_kernel<<<(NB * NBATCH) / 8, 256, 0, stream>>>(Ubuf, UF);
  bpr_wmma_kernel<<<(NB * NBATCH) / 16, 32, 0, stream>>>(UF, IPN, acc);

  // regularization
  sqsum_kernel<<<1024, 256, 0, stream>>>(ue, (long)(N_USERS + 1) * DIMS, acc + 1);
  sqsum_kernel<<<1024, 256, 0, stream>>>(ie, (long)(N_ITEMS + 1) * DIMS, acc + 2);
  finalize_kernel<<<1, 1, 0, stream>>>(acc, out);
}